// Ransac_67130338836616
// MI455X (gfx1250) — compile-verified
//
#include <hip/hip_runtime.h>

// Plane fit z = a*x + b*y + c over N = 8,388,608 float32 points.
// Normal equations via the 4x4 Gram of G[p] = [x, y, z, 1], accumulated in
// full fp32 with V_WMMA_F32_16X16X4_F32 (the accumulator operand carries the
// running Gram). Problem is HBM-bound: ~100 MB @ 23.3 TB/s ~= 4.3 us, so the
// goal is a branchless inner loop that issues ~16 instructions per 192 B.

typedef __attribute__((ext_vector_type(2))) float v2f;
typedef __attribute__((ext_vector_type(8))) float v8f;

#define NPTS          8388608                     // 2^23
#define NBLOCKS       512
#define TPB           256                         // 8 waves (wave32)
#define WAVES_PER_BLK (TPB / 32)
#define NWAVES        (NBLOCKS * WAVES_PER_BLK)   // 4096
#define PTS_PER_WAVE  (NPTS / NWAVES)             // 2048
#define PTS_PER_ITER  16                          // 4 WMMAs x K=4
#define NITERS        (PTS_PER_WAVE / PTS_PER_ITER) // 128
#define ITER_FLOATS   (PTS_PER_ITER * 3)          // 48 floats / iter / wave

static_assert(NPTS == NWAVES * PTS_PER_WAVE, "even split");
static_assert(PTS_PER_WAVE % PTS_PER_ITER == 0, "even iters");

// d_ws layout: [0, 192) ones table (48 floats); [256, 256+32768) partials.
#define WS_PARTIALS_OFF 256

__global__ void init_ones_kernel(float* __restrict__ table) {
  if (threadIdx.x < ITER_FLOATS) table[threadIdx.x] = 1.0f;
}

__global__ __launch_bounds__(TPB) void gram_wmma_kernel(
    const float* __restrict__ x, const float* __restrict__ ones_table,
    float* __restrict__ partials) {
  const int lane     = threadIdx.x & 31;
  const int waveInWg = threadIdx.x >> 5;
  const int waveId   = blockIdx.x * WAVES_PER_BLK + waveInWg;

  // f32 WMMA 16x16x4 fragment: lane L, reg r holds A[L%16][2*(L/16)+r] and
  // (mirrored) B[2*(L/16)+r][L%16]. With A[i][p] = B[p][i] = G[p][i] both
  // fragments are the same registers: value(L, r) = G[p0+2*(L/16)+r][L%16].
  //   c = L%16: 0=x 1=y 2=z 3=one; c>=4 = harmless garbage (D rows/cols >=4
  //   are never read, and each D element is computed independently).
  const int  c      = lane & 15;
  const int  h      = lane >> 4;            // half-wave -> K offset 0 / 2
  const bool is_one = (c == 3);
  const int  comp   = (c < 3) ? c : 2;      // in-bounds component for all lanes

  const long p0 = (long)waveId * PTS_PER_WAVE;
  // Lane c==3 reads 1.0f from a stride-0 table; all others stream the data.
  const float* lp = is_one ? ones_table : (x + (p0 + 2 * h) * 3 + comp);
  const long strideBytes =
      is_one ? 0 : (long)(ITER_FLOATS * sizeof(float));   // per-lane stride

  v8f acc0 = {}, acc1 = {}, acc2 = {}, acc3 = {};

  for (int it = 0; it < NITERS; ++it) {
    // 4 chunks of 4 points; reg r=0/1 -> points p+0 / p+1 within the chunk.
    v2f a0 = {lp[0],  lp[3]};
    v2f a1 = {lp[12], lp[15]};
    v2f a2 = {lp[24], lp[27]};
    v2f a3 = {lp[36], lp[39]};

    // Prefetch 8 iterations (1536 B) ahead; speculative, dropped on fault.
    __builtin_prefetch(lp + 8 * ITER_FLOATS, 0, 3);

    acc0 = __builtin_amdgcn_wmma_f32_16x16x4_f32(false, a0, false, a0,
                                                 (short)0, acc0, false, false);
    acc1 = __builtin_amdgcn_wmma_f32_16x16x4_f32(false, a1, false, a1,
                                                 (short)0, acc1, false, false);
    acc2 = __builtin_amdgcn_wmma_f32_16x16x4_f32(false, a2, false, a2,
                                                 (short)0, acc2, false, false);
    acc3 = __builtin_amdgcn_wmma_f32_16x16x4_f32(false, a3, false, a3,
                                                 (short)0, acc3, false, false);

    // Branchless per-lane advance (0 for the ones-lane) -> v_add_nc_u64.
    lp = (const float*)((const char*)lp + strideBytes);
  }

  acc0 = acc0 + acc1;
  acc2 = acc2 + acc3;
  acc0 = acc0 + acc2;

  // Gram[m][n] (m,n<4) lives at accumulator VGPR m, lane n.
  __shared__ float lds[WAVES_PER_BLK * 16];
  if (lane < 4) {
#pragma unroll
    for (int m = 0; m < 4; ++m)
      lds[waveInWg * 16 + m * 4 + lane] = acc0[m];
  }
  __syncthreads();

  if (threadIdx.x < 16) {
    float s = 0.0f;
#pragma unroll
    for (int w = 0; w < WAVES_PER_BLK; ++w) s += lds[w * 16 + threadIdx.x];
    partials[blockIdx.x * 16 + threadIdx.x] = s;   // overwrite: replay-safe
  }
}

__global__ __launch_bounds__(TPB) void reduce_solve_kernel(
    const float* __restrict__ partials, float* __restrict__ out) {
  __shared__ float lds[16 * 16];
  const int t = threadIdx.x;
  const int e = t & 15;      // Gram element index m*4+n
  const int r = t >> 4;

  float s = 0.0f;
  for (int g = r; g < NBLOCKS; g += 16) s += partials[g * 16 + e];
  lds[r * 16 + e] = s;
  __syncthreads();

  if (t < 16) {
    float tot = 0.0f;
#pragma unroll
    for (int rr = 0; rr < 16; ++rr) tot += lds[rr * 16 + e];
    lds[e] = tot;            // thread e owns column e: race-free
  }
  __syncthreads();

  if (t == 0) {
    // G = [x, y, z, 1]; element e = m*4+n.
    double a11 = lds[0];     // Sxx
    double a12 = lds[1];     // Sxy
    double b1  = lds[2];     // Sxz
    double a13 = lds[3];     // Sx
    double a22 = lds[5];     // Syy
    double b2  = lds[6];     // Syz
    double a23 = lds[7];     // Sy
    double b3  = lds[11];    // Sz
    double a33 = lds[15];    // N

    // Symmetric 3x3 solve via adjugate (f64 for conditioning headroom).
    double c11 = a22 * a33 - a23 * a23;
    double c12 = a13 * a23 - a12 * a33;
    double c13 = a12 * a23 - a13 * a22;
    double c22 = a11 * a33 - a13 * a13;
    double c23 = a12 * a13 - a11 * a23;
    double c33 = a11 * a22 - a12 * a12;
    double det = a11 * c11 + a12 * c12 + a13 * c13;
    double inv = 1.0 / det;

    out[0] = (float)((c11 * b1 + c12 * b2 + c13 * b3) * inv);  // a
    out[1] = (float)((c12 * b1 + c22 * b2 + c23 * b3) * inv);  // b
    out[2] = (float)((c13 * b1 + c23 * b2 + c33 * b3) * inv);  // c
  }
}

extern "C" void kernel_launch(void* const* d_in, const int* in_sizes, int n_in,
                              void* d_out, int out_size, void* d_ws, size_t ws_size,
                              hipStream_t stream) {
  const float* x  = (const float*)d_in[0];                       // (N, 3) f32
  float* table    = (float*)d_ws;                                // 48 floats
  float* partials = (float*)((char*)d_ws + WS_PARTIALS_OFF);     // 512*16 f32
  float* out      = (float*)d_out;                               // 3 floats

  init_ones_kernel<<<1, 64, 0, stream>>>(table);
  gram_wmma_kernel<<<NBLOCKS, TPB, 0, stream>>>(x, table, partials);
  reduce_solve_kernel<<<1, TPB, 0, stream>>>(partials, out);
}